// GaussianizationLayer_16810501997245
// MI455X (gfx1250) — compile-verified
//
#include <hip/hip_runtime.h>
#include <hip/hip_bf16.h>
#include <stdint.h>

// ---------------------------------------------------------------------------
// GaussianizationLayer: per-element bisection on a logistic-KDE inverse CDF.
//   inputs : u[2048,64] f32, datapoints[100,64] f32, log_hs[100,64] f32
//   outputs: root[2048,64] f32  ++  (-logd)[2048] f32   (concatenated)
// Compute-bound on transcendentals; memory traffic is negligible.
// ---------------------------------------------------------------------------

#define GL_DIM   64
#define GL_NDP   100
#define GL_NHD   (GL_NDP * GL_DIM)

constexpr float GL_TOL          = 1e-6f;
constexpr int   GL_MAX_IT       = 1000;
constexpr float GL_CLO          = 5e-8f;
constexpr float GL_CHI          = 1.0f - 5e-8f;     // rounds to 0x3F7FFFFF
constexpr float GL_LOG_N        = 4.6051701859880914f;  // log(100)
constexpr float GL_HALF_LOG_2PI = 0.91893853320467274f;
constexpr float GL_S_CLAMP      = 60.0f;

// ---- optional CDNA5 async global->LDS staging ------------------------------
#if defined(__has_builtin)
#  if __has_builtin(__builtin_amdgcn_global_load_async_to_lds_b32)
#    define GL_HAVE_ASYNC_LDS 1
#  endif
#endif
#ifndef GL_HAVE_ASYNC_LDS
#  define GL_HAVE_ASYNC_LDS 0
#endif

#if GL_HAVE_ASYNC_LDS
// Builtin signature (from hipcc diagnostic): first param is addrspace(1) int*,
// LDS destination is addrspace(3) int*; args = (gaddr, ldsaddr, offset, cpol).
typedef __attribute__((address_space(1))) int gl_gbl_i32;
typedef __attribute__((address_space(3))) int gl_lds_i32;
#endif

struct GlEval {
    float cdf;    // mean_n sigmoid(s_n)
    float lcdf;   // log cdf
    float lccdf;  // log (1-cdf), stable in tail
    float lpdf;   // log pdf (only when requested)
};

// One pass over the N=100 datapoints for column d.
// sA[n*64+d] = 1/h_{n,d}, sB[n*64+d] = datapoint_{n,d}/h_{n,d}.
// s clamped to +-60 so every fp32 intermediate stays in normal range.
template <bool WANT_PDF>
__device__ __forceinline__ GlEval gl_kde(float x, int d,
                                         const float* __restrict__ sA,
                                         const float* __restrict__ sB) {
    float a1 = 0.0f, a2 = 0.0f, a3 = 0.0f;
#pragma unroll 4
    for (int n = 0; n < GL_NDP; ++n) {
        float rh = sA[n * GL_DIM + d];
        float b2 = sB[n * GL_DIM + d];
        float s  = __builtin_fmaf(x, rh, -b2);
        s        = fminf(fmaxf(s, -GL_S_CLAMP), GL_S_CLAMP);
        float e  = __expf(-s);                       // v_exp_f32 path
        float sg = __builtin_amdgcn_rcpf(1.0f + e);  // sigmoid(s)
        float cs = e * sg;                           // sigmoid(-s), exact in tail
        a1 += sg;
        a2 += cs;
        if (WANT_PDF) a3 = __builtin_fmaf(cs * sg, rh, a3);
    }
    GlEval r;
    r.cdf   = a1 * 0.01f;
    r.lcdf  = __logf(a1) - GL_LOG_N;
    r.lccdf = __logf(a2) - GL_LOG_N;
    r.lpdf  = WANT_PDF ? (__logf(a3) - GL_LOG_N) : 0.0f;
    return r;
}

__device__ __forceinline__ float gl_inverse(const GlEval& r) {
    if (r.cdf >= GL_CHI) return sqrtf(-2.0f * r.lccdf);    // right tail (m_right)
    if (r.cdf <= GL_CLO) return -sqrtf(-2.0f * r.lcdf);    // left tail  (m_left)
    return normcdfinvf(r.cdf);                             // good region (ndtri)
}

// ---------------------------------------------------------------------------
// Kernel 1: 256 threads = 4 batch rows x 64 dims; grid = BATCH/4.
// Stages 1/h and d/h into LDS (async-to-LDS when available), then per-element
// bisection, final evaluation, writes root and the per-element logd term.
// ---------------------------------------------------------------------------
extern "C" __global__ void __launch_bounds__(256)
gl_bisect_kernel(const float* __restrict__ u,
                 const float* __restrict__ dp,
                 const float* __restrict__ lhs,
                 float* __restrict__ out_root,
                 float* __restrict__ tbuf) {
    __shared__ float sA[GL_NHD];   // staged datapoints -> then 1/h
    __shared__ float sB[GL_NHD];   // staged log_hs     -> then d/h
    const int tid = threadIdx.x;

#if GL_HAVE_ASYNC_LDS
    for (int i = tid; i < GL_NHD; i += 256) {
        gl_gbl_i32* g0 = (gl_gbl_i32*)(uintptr_t)(dp + i);
        gl_gbl_i32* g1 = (gl_gbl_i32*)(uintptr_t)(lhs + i);
        gl_lds_i32* l0 = (gl_lds_i32*)(unsigned)(uintptr_t)(sA + i);
        gl_lds_i32* l1 = (gl_lds_i32*)(unsigned)(uintptr_t)(sB + i);
        __builtin_amdgcn_global_load_async_to_lds_b32(g0, l0, 0, 0);
        __builtin_amdgcn_global_load_async_to_lds_b32(g1, l1, 0, 0);
    }
#  if __has_builtin(__builtin_amdgcn_s_wait_asynccnt)
    __builtin_amdgcn_s_wait_asynccnt(0);
#  else
    asm volatile("s_wait_asynccnt 0" ::: "memory");
#  endif
#else
    for (int i = tid; i < GL_NHD; i += 256) {
        sA[i] = dp[i];
        sB[i] = lhs[i];
    }
#endif
    __syncthreads();

    // transform in place: sA <- 1/max(exp(log_hs),1e-7), sB <- dp * (1/h)
    for (int i = tid; i < GL_NHD; i += 256) {
        float h  = fmaxf(__expf(sB[i]), 1e-7f);
        float rh = 1.0f / h;            // once per block: exact division is fine
        float b2 = sA[i] * rh;
        sA[i] = rh;
        sB[i] = b2;
    }
    __syncthreads();

    const int d = tid & (GL_DIM - 1);
    const int b = blockIdx.x * 4 + (tid >> 6);
    const float uu = u[b * GL_DIM + d];

    float lo = -10000.0f, hi = 10000.0f;
    for (int it = 0; it < GL_MAX_IT; ++it) {
        if (!((hi - lo) >= GL_TOL)) break;
        float mid = 0.5f * (lo + hi);
        GlEval e  = gl_kde<false>(mid, d, sA, sB);
        float  v  = gl_inverse(e);
        if (v < uu) lo = mid;
        if (v > uu) hi = mid;
        if (!(v < uu) && !(v > uu)) break;   // exact hit / NaN: interval frozen
    }

    const float root = 0.5f * (lo + hi);
    GlEval e = gl_kde<true>(root, d, sA, sB);

    float log_g;
    if (e.cdf >= GL_CHI) {
        log_g = 0.5f * __logf(-2.0f * e.lccdf) - e.lccdf;          // logd_right
    } else if (e.cdf <= GL_CLO) {
        log_g = 0.5f * __logf(-2.0f * e.lcdf) - e.lcdf;            // logd_left
    } else {
        float inv = normcdfinvf(e.cdf);
        log_g = -0.5f * inv * inv - GL_HALF_LOG_2PI;               // logd_good
    }

    out_root[b * GL_DIM + d] = root;
    tbuf[b * GL_DIM + d]     = e.lpdf - log_g;   // per-element logd term
}

// ---------------------------------------------------------------------------
// Kernel 2: row-sum of tbuf[b][0..63] via V_WMMA_F32_16X16X4_F32.
// Per wave32: 16 rows. A = 16x4 f32 tile of t; B = e0 selector (B[k][0]=1).
// 16 chained WMMAs accumulate D[:,0] = sum_k t[row][k].  C-layout: lane 0
// holds D[0..7][0] in c[0..7], lane 16 holds D[8..15][0].
// ---------------------------------------------------------------------------
typedef float gl_v2f __attribute__((ext_vector_type(2)));
typedef float gl_v8f __attribute__((ext_vector_type(8)));

extern "C" __global__ void __launch_bounds__(128)
gl_reduce_kernel(const float* __restrict__ tbuf, float* __restrict__ out_logd) {
    const int lane    = threadIdx.x & 31;
    const int wid     = threadIdx.x >> 5;
    const int rowbase = (blockIdx.x * 4 + wid) * 16;
    const int m       = lane & 15;   // M row of A, N column of B and D held by this lane
    const int kh      = lane >> 4;   // 0 -> K{0,1}, 1 -> K{2,3} of each chunk

    gl_v2f bfrag;
    bfrag.x = (m == 0) ? 1.0f : 0.0f;   // B[k][n] = (n==0)
    bfrag.y = bfrag.x;

    gl_v8f c = {};
    const float* rowp = tbuf + (rowbase + m) * GL_DIM;
#pragma unroll
    for (int kk = 0; kk < 16; ++kk) {
        const int k0 = kk * 4 + kh * 2;
        gl_v2f afrag;
        afrag.x = rowp[k0];
        afrag.y = rowp[k0 + 1];
        // (neg_a, A, neg_b, B, c_mod, C, reuse_a, reuse_b)
        c = __builtin_amdgcn_wmma_f32_16x16x4_f32(false, afrag, false, bfrag,
                                                  (short)0, c, false, false);
    }

    if (m == 0) {
#pragma unroll
        for (int j = 0; j < 8; ++j)
            out_logd[rowbase + kh * 8 + j] = -c[j];   // reference returns -logd
    }
}

// ---------------------------------------------------------------------------
extern "C" void kernel_launch(void* const* d_in, const int* in_sizes, int n_in,
                              void* d_out, int out_size, void* d_ws, size_t ws_size,
                              hipStream_t stream) {
    const float* u   = (const float*)d_in[0];
    const float* dp  = (const float*)d_in[1];
    const float* lhs = (const float*)d_in[2];
    float* out  = (float*)d_out;
    float* tbuf = (float*)d_ws;                 // BATCH*DIM floats of scratch

    const int batch = in_sizes[0] / GL_DIM;     // 2048

    gl_bisect_kernel<<<batch / 4, 256, 0, stream>>>(u, dp, lhs, out, tbuf);
    gl_reduce_kernel<<<batch / 64, 128, 0, stream>>>(tbuf, out + batch * GL_DIM);
}